// Feature2SemanticBlock_67284957659470
// MI455X (gfx1250) — compile-verified
//
#include <hip/hip_runtime.h>
#include <math.h>

// ---------------- problem constants ----------------
#define Bn   16
#define CLSn 20
#define Cn   512
#define Hn   4
#define HPn  32
#define WPn  32
#define Kn   9
#define Gn   80          // H*CLS
#define Nn   1024        // HP*WP
#define HDn  128         // C/H
#define EPSF 1e-5f
#define BNSF 0.9999950000374997f      // 1/sqrt(1+1e-5)
#define INV_SQRT2 0.7071067811865475f
#define ATT_SCALE 0.0883883476483184f // 128^-0.5
#define KV_PAD 32768                  // pad so OOB N-tile reads stay mapped

typedef __attribute__((ext_vector_type(2))) float v2f;
typedef __attribute__((ext_vector_type(8))) float v8f;

// =====================================================================
// Generic batched, strided fp32 WMMA GEMM (V_WMMA_F32_16X16X4_F32):
//   D[m,n] = post( alpha * sum_k A[m,k]*B[k,n] )
//   post: +bias[m]; then *gamma[m]*BNS + beta[m]; then + resid[m,n]
// One wave -> 16(M) x 64(N) tile, 4 accumulators sharing the A fragment.
// CONTRACT (guaranteed by the launcher):
//   * M multiple of 16, Kd multiple of 4
//   * B reads for cols [0, tilesN*64) hit mapped memory (pad buffers);
//     garbage cols are discarded by the store guard (col < Ncol).
// Inner loop is branch-free: no EXEC juggling around loads.
// =====================================================================
__global__ void wmma_gemm_f32(
    const float* __restrict__ A, const float* __restrict__ Bm,
    float* __restrict__ D,
    const float* __restrict__ bias, const float* __restrict__ gamma,
    const float* __restrict__ beta, const float* __restrict__ resid,
    float alpha, int M, int Ncol, int Kd,
    long long sam, long long sak,
    long long sbk, long long sbn,
    long long sdm, long long sdn,
    long long srm, long long srn,
    int boCount, int biCount,
    long long boA, long long biA,
    long long boB, long long biB,
    long long boD, long long biD,
    long long boR, long long biR,
    int tilesM, int tilesN)
{
    int wave = (int)((blockIdx.x * blockDim.x + threadIdx.x) >> 5);
    int totalWaves = boCount * biCount * tilesM * tilesN;
    if (wave >= totalWaves) return;            // wave-uniform guard

    int lane = threadIdx.x & 31;
    int half = lane >> 4;                      // 0 or 1
    int l16  = lane & 15;

    int tn = wave % tilesN;  int t = wave / tilesN;
    int tm = t % tilesM;     t /= tilesM;
    int bi = t % biCount;    int bo = t / biCount;

    int m0 = tm * 16, n0 = tn * 64;

    // per-lane base pointers (hoisted; inner loop only increments)
    const float* Ap = A + bo * boA + bi * biA
                        + (long long)(m0 + l16) * sam + (long long)(2 * half) * sak;
    const float* Bbase = Bm + bo * boB + bi * biB + (long long)(2 * half) * sbk;
    const float* Bp[4];
#pragma unroll
    for (int j = 0; j < 4; ++j)
        Bp[j] = Bbase + (long long)(n0 + j * 16 + l16) * sbn;

    v8f acc[4];
#pragma unroll
    for (int j = 0; j < 4; ++j) acc[j] = (v8f){0.f,0.f,0.f,0.f,0.f,0.f,0.f,0.f};

    const long long aInc = 4 * sak, bInc = 4 * sbk;

#pragma unroll 2
    for (int k = 0; k < Kd; k += 4) {
        v2f a;
        a.x = Ap[0];
        a.y = Ap[sak];
        Ap += aInc;
#pragma unroll
        for (int j = 0; j < 4; ++j) {
            v2f bf;
            bf.x = Bp[j][0];
            bf.y = Bp[j][sbk];
            Bp[j] += bInc;
            acc[j] = __builtin_amdgcn_wmma_f32_16x16x4_f32(
                         false, a, false, bf, (short)0, acc[j], false, false);
        }
    }

    float*       Dp = D + bo * boD + bi * biD;
    const float* Rp = resid ? (resid + bo * boR + bi * biR) : (const float*)0;

#pragma unroll
    for (int j = 0; j < 4; ++j) {
        int col = n0 + j * 16 + l16;
        if (col < Ncol) {
            long long dc = (long long)col * sdn;
            long long rc = (long long)col * srn;
#pragma unroll
            for (int r = 0; r < 8; ++r) {
                int row = m0 + r + 8 * half;           // always < M (M % 16 == 0)
                float v = acc[j][r] * alpha;
                if (bias)  v += bias[row];
                if (gamma) v = v * (gamma[row] * BNSF) + (beta ? beta[row] : 0.f);
                if (Rp)    v += Rp[(long long)row * srm + rc];
                Dp[(long long)row * sdm + dc] = v;
            }
        }
    }
}

// ---------------- patch global BN: per-batch mean/rstd ----------------
__global__ void patch_stats(const float* __restrict__ x, float* __restrict__ st)
{
    __shared__ float s1[256], s2[256];
    int b = blockIdx.x, tid = threadIdx.x;
    const float* xb = x + (size_t)b * Cn * Nn;
    float a = 0.f, q = 0.f;
    for (int i = tid; i < Cn * Nn; i += 256) { float v = xb[i]; a += v; q += v * v; }
    s1[tid] = a; s2[tid] = q; __syncthreads();
    for (int s = 128; s > 0; s >>= 1) {
        if (tid < s) { s1[tid] += s1[tid + s]; s2[tid] += s2[tid + s]; }
        __syncthreads();
    }
    if (tid == 0) {
        float m   = s1[0] / (float)(Cn * Nn);
        float var = s2[0] / (float)(Cn * Nn) - m * m;
        st[2 * b] = m; st[2 * b + 1] = rsqrtf(var + EPSF);
    }
}

__global__ void patch_apply(const float* __restrict__ x, const float* __restrict__ st,
                            const float* __restrict__ nw, const float* __restrict__ nb,
                            float* __restrict__ xn)
{
    size_t i = (size_t)blockIdx.x * blockDim.x + threadIdx.x;
    if (i >= (size_t)Bn * Cn * Nn) return;
    int b = (int)(i / ((size_t)Cn * Nn));
    int c = (int)((i / Nn) % Cn);
    xn[i] = (x[i] - st[2 * b]) * st[2 * b + 1] * nw[c] + nb[c];
}

// ---------------- cls layernorm + kv projection -----------------------
__global__ void cls_ln_kv(const float* __restrict__ xcls,
                          const float* __restrict__ nw, const float* __restrict__ nb,
                          const float* __restrict__ pw, const float* __restrict__ pb,
                          float* __restrict__ kv)
{
    __shared__ float xc[Cn];
    __shared__ float red[256];
    __shared__ float mean_s, rstd_s;
    int tok = blockIdx.x, tid = threadIdx.x;
    const float* xr = xcls + (size_t)tok * Cn;
    float a0 = xr[tid], a1 = xr[tid + 256];
    red[tid] = a0 + a1; __syncthreads();
    for (int s = 128; s > 0; s >>= 1) { if (tid < s) red[tid] += red[tid + s]; __syncthreads(); }
    if (tid == 0) mean_s = red[0] / (float)Cn;
    __syncthreads();
    float m = mean_s, d0 = a0 - m, d1 = a1 - m;
    red[tid] = d0 * d0 + d1 * d1; __syncthreads();
    for (int s = 128; s > 0; s >>= 1) { if (tid < s) red[tid] += red[tid + s]; __syncthreads(); }
    if (tid == 0) rstd_s = rsqrtf(red[0] / (float)Cn + EPSF);
    __syncthreads();
    float r = rstd_s;
    xc[tid]       = d0 * r * nw[tid]       + nb[tid];
    xc[tid + 256] = d1 * r * nw[tid + 256] + nb[tid + 256];
    __syncthreads();
    for (int o = tid; o < 2 * Cn; o += 256) {
        const float* wr = pw + (size_t)o * Cn;
        float acc = 0.f;
        for (int c = 0; c < Cn; ++c) acc += xc[c] * wr[c];
        kv[(size_t)tok * 2 * Cn + o] = acc + pb[o];
    }
}

// ---------------- point L2 norms -------------------------------------
__global__ void point_norms(const float* __restrict__ xf,
                            float* __restrict__ rn, float* __restrict__ sq)
{
    int p = blockIdx.x * blockDim.x + threadIdx.x;
    if (p >= Bn * Nn) return;
    const float* r = xf + (size_t)p * Gn;
    float s = 0.f;
    for (int g = 0; g < Gn; ++g) s += r[g] * r[g];
    float inv = 1.f / fmaxf(sqrtf(s), 1e-12f);
    rn[p] = inv;
    sq[p] = s * inv * inv;
}

// ---------------- KNN: top-K of -dist^2 (includes self) --------------
__global__ void knn_topk(const float* __restrict__ xf, const float* __restrict__ rn,
                         const float* __restrict__ sq, int* __restrict__ nn)
{
    __shared__ float qv[Gn];
    __shared__ float nd[Nn];
    __shared__ float rv[256];
    __shared__ int   ri[256];
    int p = blockIdx.x, tid = threadIdx.x;
    int b = p / Nn, n = p % Nn;
    const float* xb = xf + (size_t)b * Nn * Gn;
    const float* rb = rn + (size_t)b * Nn;
    const float* sb = sq + (size_t)b * Nn;
    if (tid < Gn) qv[tid] = xb[(size_t)n * Gn + tid] * rb[n];
    __syncthreads();
    float sqn = sb[n];
    for (int m = tid; m < Nn; m += 256) {
        const float* xm = xb + (size_t)m * Gn;
        float d = 0.f;
        for (int g = 0; g < Gn; ++g) d += qv[g] * xm[g];
        d *= rb[m];
        nd[m] = 2.f * d - sqn - sb[m];
    }
    __syncthreads();
    for (int kk = 0; kk < Kn; ++kk) {
        float best = -1e30f; int bidx = Nn;
        for (int m = tid; m < Nn; m += 256) {
            float v = nd[m];
            if (v > best || (v == best && m < bidx)) { best = v; bidx = m; }
        }
        rv[tid] = best; ri[tid] = bidx; __syncthreads();
        for (int s = 128; s > 0; s >>= 1) {
            if (tid < s) {
                float v = rv[tid + s]; int i2 = ri[tid + s];
                if (v > rv[tid] || (v == rv[tid] && i2 < ri[tid])) { rv[tid] = v; ri[tid] = i2; }
            }
            __syncthreads();
        }
        if (tid == 0) { nn[(size_t)p * Kn + kk] = ri[0]; nd[ri[0]] = -1e38f; }
        __syncthreads();
    }
}

// ---------------- edge feature + interleaved stack --------------------
__global__ void edge_y(const float* __restrict__ xf, const int* __restrict__ nn,
                       float* __restrict__ y)
{
    int p = blockIdx.x, g = threadIdx.x;
    if (g >= Gn) return;
    const float* row = xf + (size_t)p * Gn;
    const float* xb  = xf + (size_t)(p / Nn) * Nn * Gn;
    float c = row[g];
    float best = -1e30f;
    for (int k = 0; k < Kn; ++k) {
        int id = nn[(size_t)p * Kn + k];
        best = fmaxf(best, xb[(size_t)id * Gn + g] - c);
    }
    y[(size_t)p * 2 * Gn + 2 * g]     = c;
    y[(size_t)p * 2 * Gn + 2 * g + 1] = best;
}

// ---------------- grouped MR linear + BN2 + exact GELU ---------------
__global__ void mr_bn_gelu(const float* __restrict__ y, const float* __restrict__ w,
                           const float* __restrict__ bvec, const float* __restrict__ g2,
                           const float* __restrict__ b2, float* __restrict__ z)
{
    __shared__ float yr[2 * Gn];
    int p = blockIdx.x, tid = threadIdx.x;   // blockDim = 160
    yr[tid] = y[(size_t)p * 2 * Gn + tid];
    __syncthreads();
    int h = tid / 40;
    const float* wr = w + (size_t)tid * 40;
    const float* yg = yr + h * 40;
    float acc = 0.f;
    for (int i = 0; i < 40; ++i) acc += yg[i] * wr[i];
    acc += bvec[tid];
    acc = acc * (g2[tid] * BNSF) + b2[tid];
    acc = 0.5f * acc * (1.f + erff(acc * INV_SQRT2));
    z[(size_t)p * 2 * Gn + tid] = acc;
}

// ---------------- softmax over CLS=20 --------------------------------
__global__ void softmax_cls(const float* __restrict__ gout, float* __restrict__ attn2)
{
    int r = blockIdx.x * blockDim.x + threadIdx.x;
    if (r >= Bn * Hn * Nn) return;
    int n = r % Nn, h = (r / Nn) % Hn, b = r / (Nn * Hn);
    const float* gp = gout + (size_t)b * Gn * Nn + (size_t)h * CLSn * Nn + n;
    float v[CLSn], mx = -1e30f;
    for (int c = 0; c < CLSn; ++c) { v[c] = gp[(size_t)c * Nn]; mx = fmaxf(mx, v[c]); }
    float s = 0.f;
    for (int c = 0; c < CLSn; ++c) { v[c] = expf(v[c] - mx); s += v[c]; }
    float inv = 1.f / s;
    float* op = attn2 + (size_t)r * CLSn;
    for (int c = 0; c < CLSn; ++c) op[c] = v[c] * inv;
}

// =====================================================================
static void launch_gemm(hipStream_t s,
    const float* A, const float* Bm, float* D,
    const float* bias, const float* gamma, const float* beta, const float* resid,
    float alpha, int M, int Ncol, int Kd,
    long long sam, long long sak, long long sbk, long long sbn,
    long long sdm, long long sdn, long long srm, long long srn,
    int boC, int biC,
    long long boA, long long biA, long long boB, long long biB,
    long long boD, long long biD, long long boR, long long biR)
{
    int tilesM = (M + 15) / 16, tilesN = (Ncol + 63) / 64;
    long long waves = (long long)boC * biC * tilesM * tilesN;
    int grid = (int)((waves * 32 + 255) / 256);
    wmma_gemm_f32<<<dim3(grid), dim3(256), 0, s>>>(
        A, Bm, D, bias, gamma, beta, resid, alpha, M, Ncol, Kd,
        sam, sak, sbk, sbn, sdm, sdn, srm, srn,
        boC, biC, boA, biA, boB, biB, boD, biD, boR, biR, tilesM, tilesN);
}

extern "C" void kernel_launch(void* const* d_in, const int* in_sizes, int n_in,
                              void* d_out, int out_size, void* d_ws, size_t ws_size,
                              hipStream_t stream)
{
    const float* x_cls   = (const float*)d_in[0];
    const float* x_patch = (const float*)d_in[1];
    const float* norm_x_w= (const float*)d_in[2];
    const float* norm_x_b= (const float*)d_in[3];
    const float* pq_w    = (const float*)d_in[4];
    const float* pq_b    = (const float*)d_in[5];
    const float* ncls_w  = (const float*)d_in[6];
    const float* ncls_b  = (const float*)d_in[7];
    const float* pkv_w   = (const float*)d_in[8];
    const float* pkv_b   = (const float*)d_in[9];
    const float* fc1_w   = (const float*)d_in[10];
    const float* fc1_b   = (const float*)d_in[11];
    const float* bn1_g   = (const float*)d_in[12];
    const float* bn1_b   = (const float*)d_in[13];
    const float* mr_w    = (const float*)d_in[14];
    const float* mr_b    = (const float*)d_in[15];
    const float* bn2_g   = (const float*)d_in[16];
    const float* bn2_b   = (const float*)d_in[17];
    const float* fc2_w   = (const float*)d_in[18];
    const float* fc2_b   = (const float*)d_in[19];
    const float* bn3_g   = (const float*)d_in[20];
    const float* bn3_b   = (const float*)d_in[21];
    const float* proj_w  = (const float*)d_in[22];
    const float* proj_b  = (const float*)d_in[23];
    float* out = (float*)d_out;
    float* ws  = (float*)d_ws;

    // ----- workspace layout (floats); kv padded by KV_PAD so the 64-wide
    //       attention N-tile can overread mapped memory; q-region recycled
    //       for y/z/gout, xn-region recycled as attention output -----
    const size_t off_stats = 0;                                    // 32
    const size_t off_xn    = 64;                                   // B*C*N
    const size_t off_q     = off_xn   + (size_t)Bn * Cn * Nn;      // B*C*N
    const size_t off_kv    = off_q    + (size_t)Bn * Cn * Nn;      // B*CLS*2C + pad
    const size_t off_amap  = off_kv   + (size_t)Bn * CLSn * 2 * Cn + KV_PAD;
    const size_t off_xf    = off_amap + (size_t)Bn * Gn * Nn;      // B*N*G
    const size_t off_rn    = off_xf   + (size_t)Bn * Nn * Gn;      // B*N
    const size_t off_sq    = off_rn   + (size_t)Bn * Nn;           // B*N
    const size_t off_idx   = off_sq   + (size_t)Bn * Nn;           // B*N*K ints
    const size_t off_attn2 = off_idx  + (size_t)Bn * Nn * Kn;      // B*H*N*CLS

    float* stats  = ws + off_stats;
    float* xn     = ws + off_xn;
    float* q      = ws + off_q;
    float* kv     = ws + off_kv;
    float* amap   = ws + off_amap;
    float* xf     = ws + off_xf;
    float* rnv    = ws + off_rn;
    float* sqv    = ws + off_sq;
    int*   nnidx  = (int*)(ws + off_idx);
    float* attn2  = ws + off_attn2;
    float* ybuf   = q;                                 // q dead after attention
    float* zbuf   = q + (size_t)Bn * Nn * 2 * Gn;
    float* gout   = q + (size_t)2 * Bn * Nn * 2 * Gn;
    float* attnout= xn;                                // xn dead after pq

    // 1) patch global BN stats + apply
    patch_stats<<<dim3(Bn), dim3(256), 0, stream>>>(x_patch, stats);
    {
        size_t tot = (size_t)Bn * Cn * Nn;
        patch_apply<<<dim3((unsigned)((tot + 255) / 256)), dim3(256), 0, stream>>>(
            x_patch, stats, norm_x_w, norm_x_b, xn);
    }

    // 2) cls LN + kv = xc @ pkv_w^T + pkv_b
    cls_ln_kv<<<dim3(Bn * CLSn), dim3(256), 0, stream>>>(
        x_cls, ncls_w, ncls_b, pkv_w, pkv_b, kv);

    // 3) q = pq_w @ xn + pq_b      (per batch: 512x1024x512)
    launch_gemm(stream, pq_w, xn, q, pq_b, 0, 0, 0, 1.f,
                Cn, Nn, Cn,
                Cn, 1,            (long long)Nn, 1,
                (long long)Nn, 1, 0, 0,
                Bn, 1, 0, 0, (long long)Cn * Nn, 0, (long long)Cn * Nn, 0, 0, 0);

    // 4) attn = scale * q^T @ key  -> amap (B,G,N)   (per (b,h): 1024x20x128)
    launch_gemm(stream, q, kv, amap, 0, 0, 0, 0, ATT_SCALE,
                Nn, CLSn, HDn,
                1, (long long)Nn,  1, (long long)2 * Cn,
                1, (long long)Nn,  0, 0,
                Bn, Hn,
                (long long)Cn * Nn, (long long)HDn * Nn,
                (long long)CLSn * 2 * Cn, HDn,
                (long long)Gn * Nn, (long long)CLSn * Nn, 0, 0);

    // 5) xf = BN1(fc1_w @ amap + fc1_b), stored (B,N,G)   (per b: 80x1024x80)
    launch_gemm(stream, fc1_w, amap, xf, fc1_b, bn1_g, bn1_b, 0, 1.f,
                Gn, Nn, Gn,
                Gn, 1,             (long long)Nn, 1,
                1, (long long)Gn,  0, 0,
                Bn, 1, 0, 0, (long long)Gn * Nn, 0, (long long)Nn * Gn, 0, 0, 0);

    // 6) KNN graph on unit-normalized xf
    point_norms<<<dim3((Bn * Nn + 255) / 256), dim3(256), 0, stream>>>(xf, rnv, sqv);
    knn_topk<<<dim3(Bn * Nn), dim3(256), 0, stream>>>(xf, rnv, sqv, nnidx);
    edge_y<<<dim3(Bn * Nn), dim3(96), 0, stream>>>(xf, nnidx, ybuf);

    // 7) grouped MR + BN2 + GELU
    mr_bn_gelu<<<dim3(Bn * Nn), dim3(160), 0, stream>>>(
        ybuf, mr_w, mr_b, bn2_g, bn2_b, zbuf);

    // 8) gout = BN3(fc2_w @ z^T + fc2_b) + amap, stored (B,G,N)  (per b: 80x1024x160)
    launch_gemm(stream, fc2_w, zbuf, gout, fc2_b, bn3_g, bn3_b, amap, 1.f,
                Gn, Nn, 2 * Gn,
                2 * Gn, 1,          1, (long long)2 * Gn,
                (long long)Nn, 1,   (long long)Nn, 1,
                Bn, 1, 0, 0, (long long)Nn * 2 * Gn, 0,
                (long long)Gn * Nn, 0, (long long)Gn * Nn, 0);

    // 9) softmax over CLS
    softmax_cls<<<dim3((Bn * Hn * Nn + 255) / 256), dim3(256), 0, stream>>>(gout, attn2);

    // 10) attnout = attn2 @ val, stored (B,C,N)   (per (b,h): 1024x128x20)
    launch_gemm(stream, attn2, kv + Cn, attnout, 0, 0, 0, 0, 1.f,
                Nn, HDn, CLSn,
                CLSn, 1,           (long long)2 * Cn, 1,
                1, (long long)Nn,  0, 0,
                Bn, Hn,
                (long long)Hn * Nn * CLSn, (long long)Nn * CLSn,
                (long long)CLSn * 2 * Cn, HDn,
                (long long)Cn * Nn, (long long)HDn * Nn, 0, 0);

    // 11) out = x_patch + proj_w @ attnout + proj_b   (per b: 512x1024x512)
    launch_gemm(stream, proj_w, attnout, out, proj_b, 0, 0, x_patch, 1.f,
                Cn, Nn, Cn,
                Cn, 1,             (long long)Nn, 1,
                (long long)Nn, 1,  (long long)Nn, 1,
                Bn, 1, 0, 0, (long long)Cn * Nn, 0,
                (long long)Cn * Nn, 0, (long long)Cn * Nn, 0);

    (void)in_sizes; (void)n_in; (void)out_size; (void)ws_size;
}